// _ConvBlock5_43018392436868
// MI455X (gfx1250) — compile-verified
//
#include <hip/hip_runtime.h>

// ---------------------------------------------------------------------------
// Graph U-Net block: pool -> 2x graph-conv -> unpool, C=64, fp32 throughout.
// Edge phases: gather + native global_atomic_add_f32 (device scope), dest
// tables L2-resident (192MB L2 holds all coarse/fine feature arrays).
// GEMMs: V_WMMA_F32_16X16X4_F32; weights staged into LDS by the Tensor Data
// Mover (TENSOR_LOAD_TO_LDS + s_wait_tensorcnt).
// ---------------------------------------------------------------------------

#define CDIM 64
#define NCOARSE 100000   // fixed by the reference problem

typedef float v2f __attribute__((ext_vector_type(2)));
typedef float v8f __attribute__((ext_vector_type(8)));
typedef unsigned int v4u __attribute__((ext_vector_type(4)));
typedef int v4i __attribute__((ext_vector_type(4)));
typedef int v8i __attribute__((ext_vector_type(8)));

// Native fp32 atomic add, device scope (L2-resident RMW). Guarantees the
// CDNA5 fast path instead of a possible CAS loop from generic atomicAdd.
__device__ __forceinline__ void atom_add_f32(float* p, float v) {
  asm volatile("global_atomic_add_f32 %0, %1, off scope:SCOPE_DEV"
               :
               : "v"(p), "v"(v)
               : "memory");
}

// LDS byte address (AS3 offset) of a __shared__ object — same value DS ops use.
__device__ __forceinline__ unsigned lds_addr_of(float* p) {
  return (unsigned)(unsigned long long)(__attribute__((address_space(3))) float*)p;
}

// TDM: DMA a 2D 64x64 f32 tile (row-major, stride 64) from global into LDS.
// D# layout per CDNA5 ISA ch.8 (groups 2/3 zero => 2D tensor).
__device__ __forceinline__ void tdm_load_w64x64(const float* gsrc, float* lds_dst) {
  unsigned long long ga = (unsigned long long)gsrc;
  v4u g0;
  g0[0] = 1u;                                   // count=1 (valid), user mode
  g0[1] = lds_addr_of(lds_dst);                 // lds_addr (bytes)
  g0[2] = (unsigned)(ga & 0xFFFFFFFFull);       // global_addr[31:0]
  g0[3] = (unsigned)((ga >> 32) & 0x01FFFFFFull) | 0x80000000u;  // [56:32] | type=2
  v8i g1;
  g1[0] = 0x00020000;        // workgroup_mask=0, data_size=2 (4B), no pad/iter
  g1[1] = (CDIM & 0xFFFF) << 16;                // tensor_dim0 = 64 (lo16)
  g1[2] = (CDIM & 0xFFFF) << 16;                // tensor_dim1 = 64 (lo16)
  g1[3] = (CDIM & 0xFFFF) << 16;                // tile_dim0 = 64
  g1[4] = CDIM;                                 // tile_dim1 = 64, tile_dim2 = 0
  g1[5] = CDIM;                                 // tensor_dim0_stride = 64
  g1[6] = 0;
  g1[7] = 0;
  v4i gz4 = {0, 0, 0, 0};
  v8i gz8 = {0, 0, 0, 0, 0, 0, 0, 0};
  // amdgpu-toolchain (clang-23) 6-arg form: (g0, g1, g2, g3, gx, cpol)
  __builtin_amdgcn_tensor_load_to_lds(g0, g1, gz4, gz4, gz8, 0);
}

// ------------------------------ zero fill ----------------------------------
__global__ void zero_f32(float* __restrict__ p, long n) {
  long i = (long)blockIdx.x * blockDim.x + threadIdx.x;
  if (i < n) p[i] = 0.0f;
}

// --------------------- edge gather * attr -> atomic scatter ----------------
// 16 lanes per edge, float4 per lane (global_load_b128 + 4x atomic f32 add).
__global__ __launch_bounds__(256) void edge_scatter(
    const float* __restrict__ srcf, const int* __restrict__ es,
    const int* __restrict__ ed, const float* __restrict__ attr,
    float* __restrict__ dstf, int ne) {
  long t = (long)blockIdx.x * blockDim.x + threadIdx.x;
  long e = t >> 4;
  if (e >= ne) return;
  // stay ahead of the linear index/attr streams (speculative prefetch:
  // past-the-end addresses are silently dropped per ISA)
  __builtin_prefetch(es + e + 4096, 0, 0);
  __builtin_prefetch(ed + e + 4096, 0, 0);
  __builtin_prefetch(attr + e + 4096, 0, 0);

  int c0 = ((int)t & 15) << 2;
  int s = es[e];
  int d = ed[e];
  float w = attr[e];
  const float4 v = *(const float4*)(srcf + (long)s * CDIM + c0);
  float* p = dstf + (long)d * CDIM + c0;
  atom_add_f32(p + 0, v.x * w);
  atom_add_f32(p + 1, v.y * w);
  atom_add_f32(p + 2, v.z * w);
  atom_add_f32(p + 3, v.w * w);
}

// --------------- dual GEMM: base = H@Wr + b ; msg = H@Wn -------------------
// One wave = one 16-row tile of H. K=64 via 16 steps of WMMA_F32_16X16X4_F32.
// A frag (16x4 f32): lane L -> m = L&15, holds K = 2*(L>>4)+{0,1}.
// B frag (4x16 f32): lane L -> n = L&15, holds K = 2*(L>>4)+{0,1}.
// D (16x16 f32, 8 VGPRs): lane L, reg r -> row = 8*(L>>4)+r, col = L&15.
__global__ __launch_bounds__(256) void gemm_dual_wmma(
    const float* __restrict__ H, const float* __restrict__ Wr,
    const float* __restrict__ Wn, const float* __restrict__ bias,
    float* __restrict__ baseOut, float* __restrict__ msgOut, int nrows) {
  __shared__ float sWr[CDIM * CDIM];
  __shared__ float sWn[CDIM * CDIM];
  __shared__ float sB[CDIM];

  // Weights via Tensor Data Mover: wave 0 issues both DMAs and waits on
  // TENSORcnt; remaining waves jump straight to the barrier.
  if (threadIdx.x == 0) {
    tdm_load_w64x64(Wr, sWr);
    tdm_load_w64x64(Wn, sWn);
    __builtin_amdgcn_s_wait_tensorcnt(0);
  }
  if (threadIdx.x < CDIM) sB[threadIdx.x] = bias[threadIdx.x];
  __syncthreads();

  const int wave = threadIdx.x >> 5;
  const int lane = threadIdx.x & 31;
  const int m    = lane & 15;
  const int kh   = lane >> 4;  // 0 or 1: which K-pair of the 4-wide K step
  const int row0 = (blockIdx.x * 8 + wave) * 16;

  int arow = row0 + m;
  if (arow > nrows - 1) arow = nrows - 1;  // clamp loads; keep EXEC all-1s
  const float* A = H + (long)arow * CDIM;

  v8f accR[4] = {};  // h @ W_root, 4 N-tiles of 16
  v8f accN[4] = {};  // h @ W_nb

  for (int ks = 0; ks < 16; ++ks) {
    const int k0 = ks * 4 + kh * 2;
    v2f a = *(const v2f*)(A + k0);
#pragma unroll
    for (int nt = 0; nt < 4; ++nt) {
      const int n = nt * 16 + m;
      v2f br;
      br.x = sWr[k0 * CDIM + n];
      br.y = sWr[(k0 + 1) * CDIM + n];
      v2f bn;
      bn.x = sWn[k0 * CDIM + n];
      bn.y = sWn[(k0 + 1) * CDIM + n];
      accR[nt] = __builtin_amdgcn_wmma_f32_16x16x4_f32(
          false, a, false, br, (short)0, accR[nt], false, false);
      accN[nt] = __builtin_amdgcn_wmma_f32_16x16x4_f32(
          false, a, false, bn, (short)0, accN[nt], false, false);
    }
  }

  if (row0 + 16 <= nrows) {
    // full tile: wave-uniform branch, no per-element predication
#pragma unroll
    for (int nt = 0; nt < 4; ++nt) {
      const int col = nt * 16 + m;
      const float b = sB[col];
#pragma unroll
      for (int r = 0; r < 8; ++r) {
        const long orow = row0 + kh * 8 + r;
        baseOut[orow * CDIM + col] = accR[nt][r] + b;
        msgOut[orow * CDIM + col]  = accN[nt][r];
      }
    }
  } else {
#pragma unroll
    for (int nt = 0; nt < 4; ++nt) {
      const int col = nt * 16 + m;
      const float b = sB[col];
#pragma unroll
      for (int r = 0; r < 8; ++r) {
        const int orow = row0 + kh * 8 + r;
        if (orow < nrows) {
          baseOut[(long)orow * CDIM + col] = accR[nt][r] + b;
          msgOut[(long)orow * CDIM + col]  = accN[nt][r];
        }
      }
    }
  }
}

// ---------------------------------------------------------------------------
extern "C" void kernel_launch(void* const* d_in, const int* in_sizes, int n_in,
                              void* d_out, int out_size, void* d_ws, size_t ws_size,
                              hipStream_t stream) {
  const float* x          = (const float*)d_in[0];
  const float* W1r        = (const float*)d_in[1];
  const float* W1n        = (const float*)d_in[2];
  const float* b1         = (const float*)d_in[3];
  const float* W2r        = (const float*)d_in[4];
  const float* W2n        = (const float*)d_in[5];
  const float* b2         = (const float*)d_in[6];
  const int*   pool_src   = (const int*)d_in[7];
  const int*   pool_dst   = (const int*)d_in[8];
  const float* pool_attr  = (const float*)d_in[9];
  const int*   conv_src   = (const int*)d_in[10];
  const int*   conv_dst   = (const int*)d_in[11];
  const float* conv_attr  = (const float*)d_in[12];
  const int*   unpool_src = (const int*)d_in[13];
  const int*   unpool_dst = (const int*)d_in[14];
  const float* unpool_attr= (const float*)d_in[15];

  const int e_pool   = in_sizes[7];
  const int e_conv   = in_sizes[10];
  const int e_unpool = in_sizes[13];
  const int n_fine   = in_sizes[0] / CDIM;
  const int n_coarse = NCOARSE;

  const long nc_elems = (long)n_coarse * CDIM;
  const long nf_elems = (long)n_fine * CDIM;

  float* buf0 = (float*)d_ws;          // xc, later h2 base/accumulator
  float* buf1 = buf0 + nc_elems;       // neighbor-GEMM message features
  float* buf2 = buf1 + nc_elems;       // h1 base/accumulator

  const dim3 blk(256);
  const int gemm_blocks = (n_coarse + 127) / 128;  // 8 waves x 16 rows

  // 1) xc = segment_sum(x[pool_src] * pool_attr, pool_dst)
  zero_f32<<<(int)((nc_elems + 255) / 256), blk, 0, stream>>>(buf0, nc_elems);
  edge_scatter<<<(int)(((long)e_pool * 16 + 255) / 256), blk, 0, stream>>>(
      x, pool_src, pool_dst, pool_attr, buf0, e_pool);

  // 2) layer 1: base1 = xc@W1r + b1 (buf2); msg-feat = xc@W1n (buf1)
  gemm_dual_wmma<<<gemm_blocks, blk, 0, stream>>>(buf0, W1r, W1n, b1, buf2,
                                                  buf1, n_coarse);
  //    h1 (buf2) += segment_sum(msgfeat[conv_src] * conv_attr, conv_dst)
  edge_scatter<<<(int)(((long)e_conv * 16 + 255) / 256), blk, 0, stream>>>(
      buf1, conv_src, conv_dst, conv_attr, buf2, e_conv);

  // 3) layer 2: base2 = h1@W2r + b2 (buf0); msg-feat = h1@W2n (buf1)
  gemm_dual_wmma<<<gemm_blocks, blk, 0, stream>>>(buf2, W2r, W2n, b2, buf0,
                                                  buf1, n_coarse);
  edge_scatter<<<(int)(((long)e_conv * 16 + 255) / 256), blk, 0, stream>>>(
      buf1, conv_src, conv_dst, conv_attr, buf0, e_conv);

  // 4) out = segment_sum(h2[unpool_src] * unpool_attr, unpool_dst)
  float* out = (float*)d_out;
  zero_f32<<<(int)((nf_elems + 255) / 256), blk, 0, stream>>>(out, nf_elems);
  edge_scatter<<<(int)(((long)e_unpool * 16 + 255) / 256), blk, 0, stream>>>(
      buf0, unpool_src, unpool_dst, unpool_attr, out, e_unpool);
}